// WOS_55413668053457
// MI455X (gfx1250) — compile-verified
//
#include <hip/hip_runtime.h>
#include <cstdint>

#define KW 3
#define CIN 8
#define HIN 64
#define WIN 64
#define BATCH 4
#define NCH 16
#define DD 72            // CIN*KW*KW
#define MDIM 144         // 2*DD
#define HO 62
#define WO 62
#define TILE 16
#define TILESX 4         // ceil(62/16)
#define XT 18            // TILE + KW - 1
#define NTHREADS 256
#define BITER 40

// Low 32 bits of a flat pointer into __shared__ are the LDS byte offset
// (LDS aperture lives entirely in addr[63:32] on CDNA5).
static __device__ __forceinline__ unsigned ldsOff(const void* p) {
  return (unsigned)(unsigned long long)(uintptr_t)p;
}

// CDNA5 async DMA: global -> LDS, tracked by ASYNCcnt (CDNA5 ISA).
static __device__ __forceinline__ void asyncLoadF32(unsigned ldsaddr, const float* g) {
  asm volatile("global_load_async_to_lds_b32 %0, %1, off"
               :: "v"(ldsaddr), "v"((unsigned long long)(uintptr_t)g)
               : "memory");
}
static __device__ __forceinline__ void waitAsync0() {
  asm volatile("s_wait_asynccnt 0" ::: "memory");
}

__global__ __launch_bounds__(NTHREADS)
void wos_kernel(const float* __restrict__ x, const float* __restrict__ weight,
                const float* __restrict__ bias, const float* __restrict__ mask,
                float* __restrict__ out) {
  __shared__ float s_x[CIN * XT * XT];   // 2592 floats: x tile with halo
  __shared__ float s_m[MDIM];            // mask row for this channel
  __shared__ float s_w[MDIM];            // rectified lweight row
  __shared__ float s_bp[1];              // bpos scalar

  const int tid = threadIdx.x;
  const int c = blockIdx.y;
  const int b = blockIdx.z;
  const int ty0 = (blockIdx.x / TILESX) * TILE;
  const int tx0 = (blockIdx.x % TILESX) * TILE;

  // ---- async stage: x halo tile (edge-clamped; clamped lanes feed only inactive outputs)
  for (int idx = tid; idx < CIN * XT * XT; idx += NTHREADS) {
    const int ch  = idx / (XT * XT);
    const int rem = idx - ch * (XT * XT);
    const int r   = rem / XT;
    const int col = rem - r * XT;
    const int gr  = min(ty0 + r,   HIN - 1);
    const int gc  = min(tx0 + col, WIN - 1);
    asyncLoadF32(ldsOff(&s_x[idx]), x + (((b * CIN + ch) * HIN + gr) * WIN + gc));
  }
  // ---- async stage mask row; rectify weights on the VALU while the DMA flies
  if (tid < MDIM) {
    asyncLoadF32(ldsOff(&s_m[tid]), mask + c * MDIM + tid);
    const float bv = bias[c];
    const float nb = fmaxf(-bv, 0.0f) * (1.0f / (float)MDIM);
    float v;
    if (tid < DD) {
      const float w1 = weight[c * MDIM + tid];
      const float w2 = weight[c * MDIM + tid + DD];
      v = fmaxf(w1, 0.0f) + fmaxf(-w2, 0.0f);          // newP
    } else {
      const float w1 = weight[c * MDIM + tid - DD];
      const float w2 = weight[c * MDIM + tid];
      v = fmaxf(-w1, 0.0f) + fmaxf(w2, 0.0f);          // newN
    }
    s_w[tid] = v + nb;
  }
  if (tid == 0) s_bp[0] = fmaxf(bias[c], 0.0f);

  waitAsync0();
  __syncthreads();

  const int ly = tid >> 4;
  const int lx = tid & 15;

  // ---- build mx[144] = [inp, -inp] + mask, register-resident
  float mx[MDIM];
#pragma unroll
  for (int ch = 0; ch < CIN; ++ch) {
#pragma unroll
    for (int ki = 0; ki < KW; ++ki) {
#pragma unroll
      for (int kj = 0; kj < KW; ++kj) {
        const int d = (ch * KW + ki) * KW + kj;
        const float v = s_x[(ch * XT + (ly + ki)) * XT + (lx + kj)];
        mx[d]      = v + s_m[d];
        mx[d + DD] = s_m[d + DD] - v;
      }
    }
  }

  float vmin = mx[0], vmax = mx[0];
#pragma unroll
  for (int j = 1; j < MDIM; ++j) {
    vmin = fminf(vmin, mx[j]);
    vmax = fmaxf(vmax, mx[j]);
  }

  // ---- bisection on threshold t: A(t) = sum_j w[j] * [mx[j] >= t] is monotone.
  // Invariant target: A(lo) > bpos, A(hi) <= bpos. Answer = min{mx >= hi}, else vmax.
  const float bpos = s_bp[0];
  float lo = vmin - 1.0f;
  float hi = vmax + 1.0f;
  for (int it = 0; it < BITER; ++it) {
    const float t = 0.5f * (lo + hi);
    float acc = 0.0f;
#pragma unroll
    for (int j = 0; j < MDIM; ++j)
      acc += (mx[j] >= t) ? s_w[j] : 0.0f;   // LDS broadcast read of s_w[j]
    if (acc <= bpos) hi = t; else lo = t;
  }

  float y = 3.0e38f;
#pragma unroll
  for (int j = 0; j < MDIM; ++j) {
    const float cand = (mx[j] >= hi) ? mx[j] : 3.0e38f;
    y = fminf(y, cand);
  }
  if (y > 2.9e38f) y = vmax;   // nothing qualified -> clip(li,0) -> max element

  const int oy = ty0 + ly;
  const int ox = tx0 + lx;
  if (oy < HO && ox < WO) {
    // reference does (B*L, NC).reshape(B,NC,Ho,Wo): flat index = n*NC + c
    out[(((b * HO + oy) * WO + ox) * NCH) + c] = y;
  }
}

extern "C" void kernel_launch(void* const* d_in, const int* in_sizes, int n_in,
                              void* d_out, int out_size, void* d_ws, size_t ws_size,
                              hipStream_t stream) {
  (void)in_sizes; (void)n_in; (void)d_ws; (void)ws_size; (void)out_size;
  const float* x      = (const float*)d_in[0];
  const float* weight = (const float*)d_in[1];
  const float* bias   = (const float*)d_in[2];
  const float* mask   = (const float*)d_in[3];
  float* out = (float*)d_out;

  dim3 grid(TILESX * TILESX, NCH, BATCH);   // 16 tiles x 16 channels x 4 batch = 1024 blocks
  wos_kernel<<<grid, NTHREADS, 0, stream>>>(x, weight, bias, mask, out);
}